// GCN_21139829031240
// MI455X (gfx1250) — compile-verified
//
#include <hip/hip_runtime.h>

#define IN_F  128
#define HID_F 96

typedef __attribute__((ext_vector_type(2))) float v2f;
typedef __attribute__((ext_vector_type(8))) float v8f;

// ---------------- graph-prep kernels ----------------

__global__ void k_fill1(float* __restrict__ p, int n) {
  int i = blockIdx.x * blockDim.x + threadIdx.x;
  if (i < n) p[i] = 1.0f;
}

__global__ void k_deg(const int* __restrict__ dst, float* __restrict__ deg, int E) {
  int e = blockIdx.x * blockDim.x + threadIdx.x;
  if (e < E) atomicAdd(&deg[dst[e]], 1.0f);
}

__global__ void k_rsqrt(float* __restrict__ d, int n) {
  int i = blockIdx.x * blockDim.x + threadIdx.x;
  if (i < n) d[i] = rsqrtf(d[i]);
}

__global__ void k_norm(const int* __restrict__ src, const int* __restrict__ dst,
                       const float* __restrict__ dinv, float* __restrict__ norm, int E) {
  int e = blockIdx.x * blockDim.x + threadIdx.x;
  if (e < E) norm[e] = dinv[src[e]] * dinv[dst[e]];
}

__global__ void k_istd(const float* __restrict__ stdv, float* __restrict__ istd) {
  int c = threadIdx.x;  // 128 threads
  istd[c] = 1.0f / stdv[c];
}

// out[i,:] = dinv[i]^2 * t[i,:] + bias  (self-loop term + bias), N*24 float4 chunks
__global__ void k_prop_init(const float* __restrict__ t, const float* __restrict__ dinv,
                            const float* __restrict__ bias, float* __restrict__ out, int total4) {
  int idx = blockIdx.x * blockDim.x + threadIdx.x;
  if (idx >= total4) return;
  int i = idx / 24;
  int c = idx - i * 24;
  float w = dinv[i];
  float w2 = w * w;
  float4 tv = ((const float4*)t)[idx];
  float4 bv = ((const float4*)bias)[c];
  float4 o;
  o.x = w2 * tv.x + bv.x;
  o.y = w2 * tv.y + bv.y;
  o.z = w2 * tv.z + bv.z;
  o.w = w2 * tv.w + bv.w;
  ((float4*)out)[idx] = o;
}

// out[dst[e],:] += norm[e] * t[src[e],:]   (24 float4 chunks per edge; atomics resolve in L2)
__global__ void k_scatter(const float* __restrict__ t, float* __restrict__ out,
                          const int* __restrict__ srcI, const int* __restrict__ dstI,
                          const float* __restrict__ norm, int E) {
  long long idx = (long long)blockIdx.x * blockDim.x + threadIdx.x;
  int e = (int)(idx / 24);
  if (e >= E) return;
  int c = (int)(idx - (long long)e * 24) * 4;
  int s = srcI[e], d = dstI[e];
  float w = norm[e];
  const float4 v = *(const float4*)(t + (size_t)s * HID_F + c);
  float* o = out + (size_t)d * HID_F + c;
  atomicAdd(o + 0, w * v.x);
  atomicAdd(o + 1, w * v.y);
  atomicAdd(o + 2, w * v.z);
  atomicAdd(o + 3, w * v.w);
}

// d_out[i] = h[i,:] . Wout[:,1] + bout[1]
__global__ void k_outproj(const float* __restrict__ h, const float* __restrict__ Wout,
                          const float* __restrict__ bout, float* __restrict__ out, int n) {
  int i = blockIdx.x * blockDim.x + threadIdx.x;
  if (i >= n) return;
  const float* hr = h + (size_t)i * HID_F;
  float acc = bout[1];
  #pragma unroll
  for (int j = 0; j < 24; ++j) {
    float4 hv = ((const float4*)hr)[j];
    acc += hv.x * Wout[(4 * j + 0) * 2 + 1];
    acc += hv.y * Wout[(4 * j + 1) * 2 + 1];
    acc += hv.z * Wout[(4 * j + 2) * 2 + 1];
    acc += hv.w * Wout[(4 * j + 3) * 2 + 1];
  }
  out[i] = acc;
}

// ---------------- WMMA GEMM: C[64-row tile, 96] = f(A)[.,K] @ W[K,96] ----------------
// Block = 128 threads = 4 waves; wave w owns rows m0+16w..+15, 6 column tiles.
// A tile staged in LDS with stride K+4 ((K+4)%64 is 4 or 36 -> conflict-free fragment reads
// for both half-waves). W read from global (L2/WGP$-resident, pipelined by compiler).
// MODE: 0 = plain A, 1 = relu(A) (fused activation of previous layer),
//       2 = (A - mean) * istd (fused standard scaler; A = raw x).
// Uniform full-tile fast path: only the last of ceil(N/64) blocks takes guarded paths.
template <int K, int MODE>
__global__ void k_gemm_wmma(const float* __restrict__ A, const float* __restrict__ W,
                            float* __restrict__ C, const float* __restrict__ mean,
                            const float* __restrict__ istd, int nrows) {
  constexpr int AP = K + 4;
  __shared__ float sA[64 * AP];
  const int tid  = threadIdx.x;
  const int wave = tid >> 5;
  const int lane = tid & 31;
  const int l16  = lane & 15;
  const int half = lane >> 4;
  const int m0   = blockIdx.x * 64;
  const bool full = (m0 + 64 <= nrows);

  if (full) {
    for (int i = tid; i < 64 * K; i += 128) {
      int r = i / K;
      int c = i - r * K;
      float v = A[(size_t)(m0 + r) * K + c];
      if constexpr (MODE == 1) v = fmaxf(v, 0.0f);
      if constexpr (MODE == 2) v = (v - mean[c]) * istd[c];
      sA[r * AP + c] = v;
    }
  } else {
    for (int i = tid; i < 64 * K; i += 128) {
      int r = i / K;
      int c = i - r * K;
      int gr = m0 + r;
      float v = 0.0f;
      if (gr < nrows) {
        v = A[(size_t)gr * K + c];
        if constexpr (MODE == 1) v = fmaxf(v, 0.0f);
        if constexpr (MODE == 2) v = (v - mean[c]) * istd[c];
      }
      sA[r * AP + c] = v;
    }
  }
  __syncthreads();

  v8f acc[6] = {};
  const float* aRow = &sA[(wave * 16 + l16) * AP + 2 * half];

  #pragma unroll 2
  for (int k = 0; k < K; k += 4) {
    v2f av;
    av.x = aRow[k];       // A[M=l16][k + 2*half + 0]
    av.y = aRow[k + 1];   // A[M=l16][k + 2*half + 1]
    const float* wRow = &W[(size_t)(k + 2 * half) * HID_F + l16];
    #pragma unroll
    for (int t = 0; t < 6; ++t) {
      v2f bv;
      bv.x = wRow[t * 16];          // B[k + 2*half + 0][col]
      bv.y = wRow[t * 16 + HID_F];  // B[k + 2*half + 1][col]
      acc[t] = __builtin_amdgcn_wmma_f32_16x16x4_f32(
          false, av, false, bv, (short)0, acc[t], false, false);
    }
  }

  const int baseRow = m0 + wave * 16 + 8 * half;  // C VGPR r -> M = r + 8*half
  if (full) {
    #pragma unroll
    for (int t = 0; t < 6; ++t) {
      #pragma unroll
      for (int r = 0; r < 8; ++r) {
        C[(size_t)(baseRow + r) * HID_F + t * 16 + l16] = acc[t][r];
      }
    }
  } else {
    #pragma unroll
    for (int t = 0; t < 6; ++t) {
      #pragma unroll
      for (int r = 0; r < 8; ++r) {
        int gm = baseRow + r;
        if (gm < nrows) C[(size_t)gm * HID_F + t * 16 + l16] = acc[t][r];
      }
    }
  }
}

// ---------------- driver ----------------

extern "C" void kernel_launch(void* const* d_in, const int* in_sizes, int n_in,
                              void* d_out, int out_size, void* d_ws, size_t ws_size,
                              hipStream_t stream) {
  const float* x    = (const float*)d_in[0];
  const int*   ei   = (const int*)d_in[1];   // [2, E] int32 (JAX x64 disabled)
  const float* mean = (const float*)d_in[3];
  const float* stdv = (const float*)d_in[4];
  const float* W1   = (const float*)d_in[5];
  const float* b1   = (const float*)d_in[6];
  const float* W2   = (const float*)d_in[7];
  const float* b2   = (const float*)d_in[8];
  const float* W3   = (const float*)d_in[9];
  const float* b3   = (const float*)d_in[10];
  const float* Wout = (const float*)d_in[11];
  const float* bout = (const float*)d_in[12];

  const int n = in_sizes[0] / IN_F;
  const int E = in_sizes[1] / 2;
  const int* srcI = ei;
  const int* dstI = ei + E;

  float* ws   = (float*)d_ws;
  float* bufA = ws;                          // n*96  (GEMM output t = h @ W)
  float* bufB = bufA + (size_t)n * HID_F;    // n*96  (prop accumulator / next h)
  float* dinv = bufB + (size_t)n * HID_F;    // n     (deg, then rsqrt in place)
  float* norm = dinv + n;                    // E
  float* istd = norm + E;                    // 128

  const int B = 256;
  const int gN    = (n + B - 1) / B;
  const int gE    = (E + B - 1) / B;
  const int g24   = (n * 24 + B - 1) / B;          // N*96 features / float4
  const long long scat = (long long)E * 24;
  const int gScat = (int)((scat + B - 1) / B);
  const int gGemm = (n + 63) / 64;

  // graph prep + scaler reciprocal
  k_fill1<<<gN, B, 0, stream>>>(dinv, n);
  k_deg<<<gE, B, 0, stream>>>(dstI, dinv, E);
  k_rsqrt<<<gN, B, 0, stream>>>(dinv, n);
  k_norm<<<gE, B, 0, stream>>>(srcI, dstI, dinv, norm, E);
  k_istd<<<1, IN_F, 0, stream>>>(stdv, istd);

  // layer 1: scaler fused into A-staging
  k_gemm_wmma<IN_F, 2><<<gGemm, 128, 0, stream>>>(x, W1, bufA, mean, istd, n);
  k_prop_init<<<g24, B, 0, stream>>>(bufA, dinv, b1, bufB, n * 24);
  k_scatter<<<gScat, B, 0, stream>>>(bufA, bufB, srcI, dstI, norm, E);

  // layer 2: ReLU of layer-1 output fused into A-staging
  k_gemm_wmma<HID_F, 1><<<gGemm, 128, 0, stream>>>(bufB, W2, bufA, nullptr, nullptr, n);
  k_prop_init<<<g24, B, 0, stream>>>(bufA, dinv, b2, bufB, n * 24);
  k_scatter<<<gScat, B, 0, stream>>>(bufA, bufB, srcI, dstI, norm, E);

  // layer 3: ReLU of layer-2 output fused into A-staging; no activation after
  k_gemm_wmma<HID_F, 1><<<gGemm, 128, 0, stream>>>(bufB, W3, bufA, nullptr, nullptr, n);
  k_prop_init<<<g24, B, 0, stream>>>(bufA, dinv, b3, bufB, n * 24);
  k_scatter<<<gScat, B, 0, stream>>>(bufA, bufB, srcI, dstI, norm, E);

  // final projection -> column 1 only (~mask selects out[:,1])
  k_outproj<<<gN, B, 0, stream>>>(bufB, Wout, bout, (float*)d_out, n);
}